// MyNetwork_69801808495354
// MI455X (gfx1250) — compile-verified
//
#include <hip/hip_runtime.h>

typedef float v2f __attribute__((ext_vector_type(2)));
typedef float v8f __attribute__((ext_vector_type(8)));

constexpr int N_ = 8192;
constexpr int D_ = 2048;
constexpr int H_ = 512;
constexpr int O_ = 32;
constexpr int HP1 = H_ + 1;   // 513

__device__ __forceinline__ v8f wmma_f32_k4(v2f a, v2f b, v8f c) {
    // D = A(16x4, f32) * B(4x16, f32) + C(16x16, f32)
    return __builtin_amdgcn_wmma_f32_16x16x4_f32(
        /*neg_a=*/false, a, /*neg_b=*/false, b,
        /*c_mod=*/(short)0, c, /*reuse_a=*/false, /*reuse_b=*/false);
}

// ---------------------------------------------------------------------------
// Stage 1: feature = relu(x @ W1^T + b1)   (8192x2048)*(2048x512) -> (8192x512)
// 2x2 register-blocked f32 WMMA: each wave computes a 32x32 output block with
// 4 accumulators, reusing each A/B fragment twice (8 FLOP/B of load traffic).
// grid = (2, 256), block = 256 (8 waves): col block = bx*8+wave, row block = by.
// ---------------------------------------------------------------------------
__global__ __launch_bounds__(256)
void feat_gemm_kernel(const float* __restrict__ x, const float* __restrict__ W1,
                      const float* __restrict__ b1, float* __restrict__ feat) {
    const int lane = threadIdx.x & 31;
    const int wv   = threadIdx.x >> 5;
    const int row0 = blockIdx.y * 32;
    const int h0   = (blockIdx.x * 8 + wv) * 32;

    const int m    = lane & 15;          // A row / B col within tile
    const int koff = (lane >> 4) * 2;    // K sub-offset per half-wave

    const float* arow0 = x  + (size_t)(row0 + m)      * D_;
    const float* arow1 = x  + (size_t)(row0 + 16 + m) * D_;
    const float* brow0 = W1 + (size_t)(h0  + m)       * D_;  // B[k][n]=W1[h0+n][k]
    const float* brow1 = W1 + (size_t)(h0  + 16 + m)  * D_;

    v8f c00 = {}, c01 = {}, c10 = {}, c11 = {};
#pragma unroll 4
    for (int k = 0; k < D_; k += 4) {
        v2f a0 = *(const v2f*)(arow0 + k + koff);
        v2f a1 = *(const v2f*)(arow1 + k + koff);
        v2f b0 = *(const v2f*)(brow0 + k + koff);
        v2f b1v = *(const v2f*)(brow1 + k + koff);
        c00 = wmma_f32_k4(a0, b0,  c00);
        c01 = wmma_f32_k4(a0, b1v, c01);
        c10 = wmma_f32_k4(a1, b0,  c10);
        c11 = wmma_f32_k4(a1, b1v, c11);
    }

    const int   n     = lane & 15;
    const float bias0 = b1[h0 + n];
    const float bias1 = b1[h0 + 16 + n];
    const int   mb    = (lane >> 4) * 8;  // C/D: vgpr r -> M = r + 8*(lane>=16)
#pragma unroll
    for (int r = 0; r < 8; ++r) {
        float* rowp0 = feat + (size_t)(row0 + mb + r)      * H_ + h0 + n;
        float* rowp1 = feat + (size_t)(row0 + 16 + mb + r) * H_ + h0 + n;
        float v;
        v = c00[r] + bias0; rowp0[0]  = v > 0.f ? v : 0.f;
        v = c01[r] + bias1; rowp0[16] = v > 0.f ? v : 0.f;
        v = c10[r] + bias0; rowp1[0]  = v > 0.f ? v : 0.f;
        v = c11[r] + bias1; rowp1[16] = v > 0.f ? v : 0.f;
    }
}

// ---------------------------------------------------------------------------
// Stage 2: sequential Sherman-Morrison RLS, single workgroup (17 wave32s).
//   px    = P @ xi                     (P symmetric -> xi@P == px^T)
//   denom = 1 + xi.px
//   W    -= px (xi@W - yi) / denom     (W in LDS, 513x32)
//   P    -= px px^T / denom            (fused with next step's matvec)
// Final-step P update is skipped (P is not an output).
// ---------------------------------------------------------------------------
__global__ __launch_bounds__(544)
void rls_kernel(const float* __restrict__ feat, const float* __restrict__ y,
                const float* __restrict__ W0, const int* __restrict__ epoch_p,
                float* __restrict__ P, float* __restrict__ Wn) {
    __shared__ float Wl[HP1 * O_];      // 513*32 f32 = 64.1 KB (LDS, 320KB/WGP)
    __shared__ float xbuf[2][HP1];
    __shared__ float pbuf[2][HP1];
    __shared__ float e_s[O_];
    __shared__ float denom_s;

    const int tid  = threadIdx.x;
    const int lane = tid & 31;
    const int wv   = tid >> 5;          // 0..16
    const int NT   = 544;

    for (int i = tid; i < HP1 * O_; i += NT) Wl[i] = W0[i];

    const int ep = epoch_p[0];
    int S = (ep >= 13) ? 1 : (N_ >> ep);   // ceil(0.5^epoch * N), N = 2^13
    if (S < 1) S = 1;

    for (int j = tid; j < HP1; j += NT)
        xbuf[0][j] = (j < H_) ? feat[j] : 1.0f;
    __syncthreads();

    // initial px = P @ x0 (coalesced: one wave per row, 32-lane sweep)
    {
        const float* xv  = xbuf[0];
        float*       pxo = pbuf[0];
        for (int j = wv; j < HP1; j += 17) {
            const float* prow = P + (size_t)j * HP1;
            float acc = 0.f;
            for (int k = lane; k < HP1; k += 32) acc += prow[k] * xv[k];
            for (int off = 16; off > 0; off >>= 1) acc += __shfl_xor(acc, off);
            if (lane == 0) pxo[j] = acc;
        }
    }
    __syncthreads();

    int cur = 0;
    for (int t = 0; t < S; ++t) {
        const float* xi = xbuf[cur];
        const float* px = pbuf[cur];

        if (tid == 0) denom_s = 1.0f;
        if (tid < O_) e_s[tid] = 0.0f;
        __syncthreads();

        // denom = 1 + xi.px  (block reduction via LDS float atomics)
        float dd = 0.f;
        for (int j = tid; j < HP1; j += NT) dd += xi[j] * px[j];
        atomicAdd(&denom_s, dd);

        // e[o] = xi @ W[:,o]  (17 partial groups per column)
        {
            const int o = tid & 31;
            float pe = 0.f;
            for (int j = tid >> 5; j < HP1; j += 17) pe += xi[j] * Wl[j * O_ + o];
            atomicAdd(&e_s[o], pe);
        }
        __syncthreads();
        if (tid < O_) e_s[tid] -= y[(size_t)t * O_ + tid];
        __syncthreads();

        const float inv = 1.0f / denom_s;

        // W -= px (outer) e / denom
        for (int i = tid; i < HP1 * O_; i += NT) {
            const int j = i >> 5, o = i & 31;
            Wl[i] -= px[j] * e_s[o] * inv;
        }

        const bool last = (t == S - 1);
        if (!last) {
            float* xn  = xbuf[cur ^ 1];
            float* pxn = pbuf[cur ^ 1];
            for (int j = tid; j < HP1; j += NT)
                xn[j] = (j < H_) ? feat[(size_t)(t + 1) * H_ + j] : 1.0f;
            __syncthreads();
            // fused: P -= inv * px px^T, and pxn = P_new @ xn in the same pass
            for (int j = wv; j < HP1; j += 17) {
                const float pj = px[j] * inv;
                float* prow = P + (size_t)j * HP1;
                float acc = 0.f;
                for (int k = lane; k < HP1; k += 32) {
                    float v = prow[k] - pj * px[k];
                    prow[k] = v;
                    acc += v * xn[k];
                }
                for (int off = 16; off > 0; off >>= 1) acc += __shfl_xor(acc, off);
                if (lane == 0) pxn[j] = acc;
            }
            cur ^= 1;
        }
        __syncthreads();
    }

    for (int i = tid; i < HP1 * O_; i += NT) Wn[i] = Wl[i];
}

// ---------------------------------------------------------------------------
// Stage 3: out = feature @ Wn[:512] + Wn[512]  (8192x512)*(513x32) -> (8192x32)
// 512 row tiles x 2 col tiles; one 16x16 tile per wave; bias row folded into C.
// Memory-bound (16.8 MB feature read), so no further blocking needed.
// ---------------------------------------------------------------------------
__global__ __launch_bounds__(256)
void out_gemm_kernel(const float* __restrict__ feat, const float* __restrict__ Wn,
                     float* __restrict__ out) {
    const int lane = threadIdx.x & 31;
    const int wv   = threadIdx.x >> 5;
    const int tile = blockIdx.x * 8 + wv;     // 0..1023
    const int row0 = (tile >> 1) * 16;
    const int n0   = (tile & 1) * 16;

    const int m    = lane & 15;
    const int koff = (lane >> 4) * 2;
    const int n    = lane & 15;

    const float* arow = feat + (size_t)(row0 + m) * H_;

    const float bias = Wn[(size_t)H_ * O_ + n0 + n];   // ones-column row of W
    v8f c;
#pragma unroll
    for (int r = 0; r < 8; ++r) c[r] = bias;

#pragma unroll 8
    for (int k = 0; k < H_; k += 4) {
        v2f a = *(const v2f*)(arow + k + koff);
        v2f b;
        b.x = Wn[(size_t)(k + koff)     * O_ + n0 + n];
        b.y = Wn[(size_t)(k + koff + 1) * O_ + n0 + n];
        c = wmma_f32_k4(a, b, c);
    }

    const int mb = (lane >> 4) * 8;
#pragma unroll
    for (int r = 0; r < 8; ++r)
        out[(size_t)(row0 + mb + r) * O_ + n0 + n] = c[r];
}

// ---------------------------------------------------------------------------
extern "C" void kernel_launch(void* const* d_in, const int* in_sizes, int n_in,
                              void* d_out, int out_size, void* d_ws, size_t ws_size,
                              hipStream_t stream) {
    const float* x     = (const float*)d_in[0];   // (8192, 2048)
    const float* y     = (const float*)d_in[1];   // (8192, 32)
    const float* W1    = (const float*)d_in[2];   // (512, 2048)
    const float* b1    = (const float*)d_in[3];   // (512,)
    const float* W0    = (const float*)d_in[4];   // (513, 32)
    const float* P0    = (const float*)d_in[5];   // (513, 513)
    const int*   epoch = (const int*)d_in[6];     // scalar
    float* out = (float*)d_out;                   // (8192, 32)

    char* ws = (char*)d_ws;
    const size_t feat_bytes = (size_t)N_ * H_ * sizeof(float);              // 16 MB
    const size_t p_bytes    = ((size_t)HP1 * HP1 * sizeof(float) + 255) & ~(size_t)255;
    float* feat = (float*)ws;
    float* Pw   = (float*)(ws + feat_bytes);
    float* Wn   = (float*)(ws + feat_bytes + p_bytes);

    // working copy of P (inputs must not be mutated); capture-safe async D2D
    hipMemcpyAsync(Pw, P0, (size_t)HP1 * HP1 * sizeof(float),
                   hipMemcpyDeviceToDevice, stream);

    feat_gemm_kernel<<<dim3(2, 256), 256, 0, stream>>>(x, W1, b1, feat);
    rls_kernel<<<1, 544, 0, stream>>>(feat, y, W0, epoch, Pw, Wn);
    out_gemm_kernel<<<128, 256, 0, stream>>>(feat, Wn, out);
}